// TimeSeriesConcentrationSNN_2010044695299
// MI455X (gfx1250) — compile-verified
//
#include <hip/hip_runtime.h>

// CDNA5 / gfx1250 fused SNN kernel.
// D = A(16x4, f32) x B(4x16, f32) + C(16x16, f32) via V_WMMA_F32_16X16X4_F32.
typedef __attribute__((ext_vector_type(2))) float v2f;
typedef __attribute__((ext_vector_type(8))) float v8f;

#define T_STEPS 200
#define BATCH   16384
#define H1      28
#define H2      14
#define ROWSTR  36      // LDS row stride in dwords (16B-aligned rows, spreads banks)
#define WPB     4       // waves per block (blockDim = 128)

__device__ __forceinline__ v8f wmma_f32(v2f a, v2f b, v8f c) {
  // 8 args: (neg_a, A, neg_b, B, c_mod, C, reuse_a, reuse_b)
  return __builtin_amdgcn_wmma_f32_16x16x4_f32(false, a, false, b, (short)0, c,
                                               false, false);
}

__global__ __launch_bounds__(128) void snn_lif_wmma_kernel(
    const float* __restrict__ x,
    const float* __restrict__ W1, const float* __restrict__ b1,
    const float* __restrict__ W2, const float* __restrict__ b2,
    const float* __restrict__ W3, const float* __restrict__ b3,
    float* __restrict__ out)
{
  __shared__ float lds[WPB][16 * ROWSTR];

  const int tid  = threadIdx.x;
  const int wave = tid >> 5;
  const int lane = tid & 31;
  const int ln   = lane & 15;   // N (or M / row) index within half-wave
  const int h    = lane >> 4;   // half-wave select
  const int b0   = (blockIdx.x * WPB + wave) * 16;  // batch-tile base row
  float* Ls = lds[wave];

  // ---------------- preload weights into matrix-register layouts ------------
  // B-matrix layout (4x16 f32): lane -> N = ln, component v & half h -> K = 2h+v.
  // B[k][n] = W[n][k] (since cur = x @ W.T).
  v2f B1t[2][2];  // [ntile][kchunk]
  #pragma unroll
  for (int nt = 0; nt < 2; ++nt) {
    #pragma unroll
    for (int c = 0; c < 2; ++c) {
      const int n = ln + 16 * nt;
      const int k = 4 * c + 2 * h;
      B1t[nt][c][0] = (n < H1) ? W1[n * 8 + k]     : 0.0f;
      B1t[nt][c][1] = (n < H1) ? W1[n * 8 + k + 1] : 0.0f;
    }
  }
  v2f B2t[7];     // K = 28 -> 7 chunks of 4
  #pragma unroll
  for (int c = 0; c < 7; ++c) {
    const int k = 4 * c + 2 * h;
    B2t[c][0] = (ln < H2) ? W2[ln * H1 + k]     : 0.0f;
    B2t[c][1] = (ln < H2) ? W2[ln * H1 + k + 1] : 0.0f;
  }
  v8f bias1[2], bias2;
  {
    const float v0 = (ln      < H1) ? b1[ln]      : 0.0f;
    const float v1 = (ln + 16 < H1) ? b1[ln + 16] : 0.0f;
    const float v2 = (ln < H2) ? b2[ln] : 0.0f;
    #pragma unroll
    for (int i = 0; i < 8; ++i) { bias1[0][i] = v0; bias1[1][i] = v1; bias2[i] = v2; }
  }
  float w3r[H2];
  #pragma unroll
  for (int j = 0; j < H2; ++j) w3r[j] = W3[j];
  const float bias3 = b3[0];

  // ---------------- persistent LIF membrane state (C layout) ----------------
  v8f m1a, m1b, m2;
  #pragma unroll
  for (int i = 0; i < 8; ++i) { m1a[i] = 0.0f; m1b[i] = 0.0f; m2[i] = 0.0f; }
  float m3 = 0.0f;

  float* out_m3 = out;
  float* out_s1 = out + (size_t)T_STEPS * BATCH;
  float* out_s2 = out + (size_t)T_STEPS * BATCH * (size_t)(1 + H1);

  for (int t = 0; t < T_STEPS; ++t) {
    // ---- layer 1: load x tile directly in A layout (lane=M, K=4c+2h+{0,1})
    const float* xr = x + ((size_t)t * BATCH + b0 + ln) * 8 + 2 * h;
    v2f a0; a0[0] = xr[0]; a0[1] = xr[1];
    v2f a1; a1[0] = xr[4]; a1[1] = xr[5];

    v8f c1a = wmma_f32(a0, B1t[0][0], bias1[0]);
    c1a     = wmma_f32(a1, B1t[0][1], c1a);
    v8f c1b = wmma_f32(a0, B1t[1][0], bias1[1]);
    c1b     = wmma_f32(a1, B1t[1][1], c1b);

    // ---- LIF layer 1 (elementwise on C-layout accumulators)
    v8f s1a, s1b;
    #pragma unroll
    for (int i = 0; i < 8; ++i) {
      float mo = m1a[i];
      float mn = __builtin_fmaf(0.9f, mo, c1a[i]) - ((mo > 1.0f) ? 1.0f : 0.0f);
      s1a[i] = (mn > 1.0f) ? 1.0f : 0.0f;  m1a[i] = mn;
      mo = m1b[i];
      mn = __builtin_fmaf(0.9f, mo, c1b[i]) - ((mo > 1.0f) ? 1.0f : 0.0f);
      s1b[i] = (mn > 1.0f) ? 1.0f : 0.0f;  m1b[i] = mn;
    }

    // ---- store spk1: C layout -> rows M=v+8h, cols N=ln(+16); lane-contiguous
    float* p1 = out_s1 + ((size_t)t * BATCH + b0 + 8 * h) * H1 + ln;
    #pragma unroll
    for (int v = 0; v < 8; ++v) p1[v * H1] = s1a[v];
    if (ln < H1 - 16) {
      #pragma unroll
      for (int v = 0; v < 8; ++v) p1[v * H1 + 16] = s1b[v];
    }

    // ---- spk1 -> LDS tile (row-major 16x28, stride ROWSTR) for A-transpose
    #pragma unroll
    for (int v = 0; v < 8; ++v) Ls[(8 * h + v) * ROWSTR + ln] = s1a[v];
    if (ln < H1 - 16) {
      #pragma unroll
      for (int v = 0; v < 8; ++v) Ls[(8 * h + v) * ROWSTR + 16 + ln] = s1b[v];
    }
    asm volatile("" ::: "memory");  // same-wave DS ops are in-order in HW

    // ---- layer 2: 7 K-chunk WMMAs, A read from LDS in A layout
    v8f c2 = bias2;
    #pragma unroll
    for (int c = 0; c < 7; ++c) {
      v2f a;
      a[0] = Ls[ln * ROWSTR + 4 * c + 2 * h];
      a[1] = Ls[ln * ROWSTR + 4 * c + 2 * h + 1];
      c2 = wmma_f32(a, B2t[c], c2);
    }
    asm volatile("" ::: "memory");

    // ---- LIF layer 2
    v8f s2;
    #pragma unroll
    for (int i = 0; i < 8; ++i) {
      float mo = m2[i];
      float mn = __builtin_fmaf(0.9f, mo, c2[i]) - ((mo > 1.0f) ? 1.0f : 0.0f);
      s2[i] = (mn > 1.0f) ? 1.0f : 0.0f;  m2[i] = mn;
    }

    // ---- store spk2 (N < 14) and stage it in LDS for the layer-3 dot
    float* p2 = out_s2 + ((size_t)t * BATCH + b0 + 8 * h) * H2 + ln;
    if (ln < H2) {
      #pragma unroll
      for (int v = 0; v < 8; ++v) p2[v * H2] = s2[v];
      #pragma unroll
      for (int v = 0; v < 8; ++v) Ls[(8 * h + v) * ROWSTR + ln] = s2[v];
    }
    asm volatile("" ::: "memory");

    // ---- layer 3: per-lane 14-wide dot (lane m owns batch row m) + LIF
    float cur3 = bias3;
    #pragma unroll
    for (int j = 0; j < H2; ++j)
      cur3 = __builtin_fmaf(w3r[j], Ls[ln * ROWSTR + j], cur3);
    {
      const float mo = m3;
      m3 = __builtin_fmaf(0.9f, mo, cur3) - ((mo > 1.0f) ? 1.0f : 0.0f);
    }
    if (lane < 16) out_m3[(size_t)t * BATCH + b0 + ln] = m3;
    asm volatile("" ::: "memory");  // fence before next step's spk1 LDS writes
  }
}

extern "C" void kernel_launch(void* const* d_in, const int* in_sizes, int n_in,
                              void* d_out, int out_size, void* d_ws, size_t ws_size,
                              hipStream_t stream) {
  (void)in_sizes; (void)n_in; (void)d_ws; (void)ws_size; (void)out_size;
  const float* x  = (const float*)d_in[0];
  const float* W1 = (const float*)d_in[1];
  const float* b1 = (const float*)d_in[2];
  const float* W2 = (const float*)d_in[3];
  const float* b2 = (const float*)d_in[4];
  const float* W3 = (const float*)d_in[5];
  const float* b3 = (const float*)d_in[6];
  float* out = (float*)d_out;

  const int blocks = BATCH / (16 * WPB);  // 256 blocks x 128 threads = 1024 waves
  snn_lif_wmma_kernel<<<blocks, 128, 0, stream>>>(x, W1, b1, W2, b2, W3, b3, out);
}